// DifferentiableMaxMedian_5145370820911
// MI455X (gfx1250) — compile-verified
//
#include <hip/hip_runtime.h>
#include <stdint.h>

// DifferentiableMaxMedian fused stencil for gfx1250 (MI455X).
// B=4, C=32, H=W=256, window=5, reflect pad=2.

#define HH 256
#define WW 256
#define TILE_W 64
#define TILE_H 32
#define LW (TILE_W + 4)   // 68
#define LH (TILE_H + 4)   // 36
#define NTHREADS 256
#define ROWS_PER_THREAD (TILE_H / 4)   // 8

__device__ __forceinline__ int reflect_idx(int i, int n) {
    i = (i < 0) ? -i : i;
    i = (i >= n) ? (2 * n - 2 - i) : i;
    return i;
}

__device__ __forceinline__ float fast_exp2(float x) { return __builtin_amdgcn_exp2f(x); }
__device__ __forceinline__ float fast_rcp(float x)  { return __builtin_amdgcn_rcpf(x); }
__device__ __forceinline__ float fast_sigmoid(float v) {
    // 1 / (1 + exp(-v)) with native exp2
    return fast_rcp(1.0f + fast_exp2(-1.4426950408889634f * v));
}

__global__ __launch_bounds__(NTHREADS)
void maxmedian_kernel(const float* __restrict__ x,
                      const float* __restrict__ mix,
                      const float* __restrict__ beta_raw,
                      float* __restrict__ out)
{
    __shared__ float tile[LH * LW];

    const int tid     = threadIdx.x;
    const int plane   = blockIdx.y;            // 0..127  (b*32 + c)
    const int c       = plane & 31;
    const int tileIdx = blockIdx.x;            // 0..31
    const int tx0     = (tileIdx & 3) * TILE_W;
    const int ty0     = (tileIdx >> 2) * TILE_H;

    const float* __restrict__ xp = x + (size_t)plane * (HH * WW);

    // ---------------- async global -> LDS tile fill (reflect padding) ------
    // Each lane supplies its own global address and LDS byte offset; the
    // async copy engine moves the data without touching VGPR data paths.
    // Low 32 bits of a generic pointer to __shared__ are the LDS byte offset.
    unsigned lds_base = (unsigned)(uintptr_t)(&tile[0]);
    for (int i = tid; i < LH * LW; i += NTHREADS) {
        int r   = i / LW;
        int col = i - r * LW;
        int gy  = reflect_idx(ty0 + r - 2, HH);
        int gx  = reflect_idx(tx0 + col - 2, WW);
        unsigned long long gaddr =
            (unsigned long long)(uintptr_t)(xp + gy * WW + gx);
        unsigned lds_off = lds_base + (unsigned)(i * 4);
        asm volatile("global_load_async_to_lds_b32 %0, %1, off"
                     :: "v"(lds_off), "v"(gaddr) : "memory");
    }
    asm volatile("s_wait_asynccnt 0" ::: "memory");
    __syncthreads();

    // ---------------- scalar-ish parameters --------------------------------
    const float beta = 5.0f + 45.0f * fast_sigmoid(beta_raw[0]);
    const float nb   = -beta * 1.4426950408889634f;   // exponent scale for exp2
    const float lam  = fast_sigmoid(mix[c]);
    const float oml  = 1.0f - lam;

    // ---------------- per-thread sliding 5x5 window in registers -----------
    const int txl  = tid & 63;                  // column within tile: 0..63
    const int tyq  = tid >> 6;                  // 0..3
    const int row0 = tyq * ROWS_PER_THREAD;     // first output row (local)

    float w[5][5];
#pragma unroll
    for (int i = 0; i < 5; ++i)
#pragma unroll
        for (int j = 0; j < 5; ++j)
            w[i][j] = tile[(row0 + i) * LW + txl + j];

    float* __restrict__ outp =
        out + (size_t)plane * (HH * WW) + (size_t)(ty0 + row0) * WW + (tx0 + txl);

#pragma unroll
    for (int p = 0; p < ROWS_PER_THREAD; ++p) {
        // Pass A: sum (for mean) and max filter over the 25 taps.
        float s = 0.0f, vmax = -__builtin_inff();
#pragma unroll
        for (int i = 0; i < 5; ++i)
#pragma unroll
            for (int j = 0; j < 5; ++j) {
                float v = w[i][j];
                s += v;
                vmax = fmaxf(vmax, v);
            }
        const float cen = s * 0.04f;            // mean of 25

        // Pass B: min |p - mean| for a numerically stable softmax.
        float dmin = __builtin_inff();
#pragma unroll
        for (int i = 0; i < 5; ++i)
#pragma unroll
            for (int j = 0; j < 5; ++j)
                dmin = fminf(dmin, fabsf(w[i][j] - cen));

        // Pass C: softmax(-beta*|d|) weighted sum (soft median).
        float wsum = 0.0f, acc = 0.0f;
#pragma unroll
        for (int i = 0; i < 5; ++i)
#pragma unroll
            for (int j = 0; j < 5; ++j) {
                float v = w[i][j];
                float e = fast_exp2(nb * (fabsf(v - cen) - dmin)); // v_exp_f32
                wsum += e;
                acc = fmaf(e, v, acc);
            }
        const float med = acc * fast_rcp(wsum);

        outp[p * WW] = lam * med + oml * vmax;

        // Slide window down one row: replace the oldest row (slot p%5) with
        // LDS row row0+p+5. Order inside w[][] is irrelevant (all reductions
        // over the window are permutation-invariant).
        if (p < ROWS_PER_THREAD - 1) {
            const int slot = p % 5;
#pragma unroll
            for (int j = 0; j < 5; ++j)
                w[slot][j] = tile[(row0 + p + 5) * LW + txl + j];
        }
    }
}

extern "C" void kernel_launch(void* const* d_in, const int* in_sizes, int n_in,
                              void* d_out, int out_size, void* d_ws, size_t ws_size,
                              hipStream_t stream) {
    const float* x        = (const float*)d_in[0];   // (4,32,256,256) f32
    const float* mix      = (const float*)d_in[1];   // (1,32,1,1)     f32
    const float* beta_raw = (const float*)d_in[2];   // scalar         f32
    float* out            = (float*)d_out;           // (4,32,256,256) f32

    dim3 grid((WW / TILE_W) * (HH / TILE_H), 4 * 32);   // 32 x 128 blocks
    maxmedian_kernel<<<grid, NTHREADS, 0, stream>>>(x, mix, beta_raw, out);
}